// GCN_list_12687333392406
// MI455X (gfx1250) — compile-verified
//
#include <hip/hip_runtime.h>

typedef __attribute__((ext_vector_type(2))) float v2f;
typedef __attribute__((ext_vector_type(8))) float v8f;

#define F_DIM 128

// ---------------------------------------------------------------- utilities
__global__ void zero_kernel(float* __restrict__ p, long long n) {
    long long i = (long long)blockIdx.x * blockDim.x + threadIdx.x;
    if (i < n) p[i] = 0.0f;
}

__global__ void deg_kernel(const int* __restrict__ dst, float* __restrict__ deg, int E) {
    int e = blockIdx.x * blockDim.x + threadIdx.x;
    if (e < E) atomicAdd(&deg[dst[e]], 1.0f);
}

__global__ void invdeg_kernel(float* __restrict__ deg, int N) {
    int i = blockIdx.x * blockDim.x + threadIdx.x;
    if (i < N) {
        float d = deg[i];
        deg[i] = (d > 0.0f) ? (1.0f / d) : 0.0f;
    }
}

// Per-edge scatter: each thread handles one edge x one 4-feature chunk.
// feat rows are 512B; gather via float4, accumulate via global_atomic_add_f32.
// All feature buffers (<=25.6MB) fit in the 192MB L2 so atomics stay on-chip.
__global__ void scatter_kernel(const float* __restrict__ feat,
                               const int* __restrict__ src,
                               const int* __restrict__ dst,
                               float* __restrict__ agg, int E) {
    long long t = (long long)blockIdx.x * blockDim.x + threadIdx.x;
    int e  = (int)(t >> 5);
    if (e >= E) return;
    int fg = ((int)t & 31) * 4;
    int s = src[e], d = dst[e];
    const float4 v = *(const float4*)(feat + (size_t)s * F_DIM + fg);
    float* o = agg + (size_t)d * F_DIM + fg;
    atomicAdd(o + 0, v.x);
    atomicAdd(o + 1, v.y);
    atomicAdd(o + 2, v.z);
    atomicAdd(o + 3, v.w);
}

// ------------------------------------------------- fp32 WMMA fused SAGE GEMM
// out[m, n0:n0+16] = relu?( (agg[m,:]*invd[m]) @ Wl + bl + x[m,:] @ Wr )
// One wave = one 16x16 output tile. blockDim = (32, 8): 8 col tiles = 128 cols.
// V_WMMA_F32_16X16X4_F32 layouts (ISA 7.12.2):
//   A 16x4 : lane half (lane>>4) selects K pair {k0,k0+1} vs {k0+2,k0+3}, row = lane&15
//   B 4x16 : same K split, col = lane&15
//   C/D    : vgpr i -> row m0 + 8*half + i, col n0 + (lane&15)
__global__ void sage_gemm128_wmma(const float* __restrict__ agg,
                                  const float* __restrict__ xin,
                                  const float* __restrict__ invd,
                                  const float* __restrict__ Wl,
                                  const float* __restrict__ bl,
                                  const float* __restrict__ Wr,
                                  float* __restrict__ out,
                                  int N, int do_relu) {
    const int lane = threadIdx.x;            // 0..31
    const int n0   = threadIdx.y * 16;       // output col tile
    const int m0   = blockIdx.x * 16;        // output row tile
    const int half = lane >> 4;              // 0 | 1
    const int l    = lane & 15;

    const int  mRow   = m0 + l;
    const bool mValid = (mRow < N);
    const int  ar     = mValid ? mRow : 0;
    const float scale = mValid ? invd[ar] : 0.0f;  // fold mean-scaling into A
    const float xmask = mValid ? 1.0f : 0.0f;      // zero padded rows (keep EXEC full)

    const float* __restrict__ aggRow = agg + (size_t)ar * F_DIM;
    const float* __restrict__ xRow   = xin + (size_t)ar * F_DIM;
    const int col = n0 + l;

    v8f acc = {0.f, 0.f, 0.f, 0.f, 0.f, 0.f, 0.f, 0.f};

    for (int k0 = 0; k0 < F_DIM; k0 += 4) {
        const int ka = k0 + half * 2;        // even -> float2-aligned
        float2 am2 = *(const float2*)(aggRow + ka);
        float2 ax2 = *(const float2*)(xRow + ka);
        v2f aM; aM.x = am2.x * scale; aM.y = am2.y * scale;
        v2f aX; aX.x = ax2.x * xmask; aX.y = ax2.y * xmask;
        v2f bL; bL.x = Wl[(size_t)ka * F_DIM + col];
                bL.y = Wl[(size_t)(ka + 1) * F_DIM + col];
        v2f bR; bR.x = Wr[(size_t)ka * F_DIM + col];
                bR.y = Wr[(size_t)(ka + 1) * F_DIM + col];
        acc = __builtin_amdgcn_wmma_f32_16x16x4_f32(false, aM, false, bL,
                                                    (short)0, acc, false, false);
        acc = __builtin_amdgcn_wmma_f32_16x16x4_f32(false, aX, false, bR,
                                                    (short)0, acc, false, false);
    }

    const float bias = bl[col];
#pragma unroll
    for (int i = 0; i < 8; ++i) {
        int row = m0 + half * 8 + i;
        if (row < N) {
            float v = acc[i] + bias;
            if (do_relu) v = (v > 0.0f) ? v : 0.0f;
            out[(size_t)row * F_DIM + col] = v;
        }
    }
}

// --------------------------------------------- layer 2 (C=4, too skinny for WMMA)
__global__ void out_kernel(const float* __restrict__ agg,
                           const float* __restrict__ h2,
                           const float* __restrict__ invd,
                           const float* __restrict__ Wl,   // 128x4
                           const float* __restrict__ bl,   // 4
                           const float* __restrict__ Wr,   // 128x4
                           float* __restrict__ out0,
                           float* __restrict__ out1,
                           int N) {
    int n = blockIdx.x * blockDim.x + threadIdx.x;
    if (n >= N) return;
    float s = invd[n];
    float a0 = bl[0], a1 = bl[1], a2 = bl[2], a3 = bl[3];
    const float* a = agg + (size_t)n * F_DIM;
    const float* x = h2  + (size_t)n * F_DIM;
    for (int k = 0; k < F_DIM; ++k) {
        float m = a[k] * s;
        float v = x[k];
        const float* wl = Wl + k * 4;
        const float* wr = Wr + k * 4;
        a0 += m * wl[0] + v * wr[0];
        a1 += m * wl[1] + v * wr[1];
        a2 += m * wl[2] + v * wr[2];
        a3 += m * wl[3] + v * wr[3];
    }
    float4 r = make_float4(a0, a1, a2, a3);
    *(float4*)(out0 + (size_t)n * 4) = r;
    *(float4*)(out1 + (size_t)n * 4) = r;
}

// -------------------------------------------------------------------- driver
extern "C" void kernel_launch(void* const* d_in, const int* in_sizes, int n_in,
                              void* d_out, int out_size, void* d_ws, size_t ws_size,
                              hipStream_t stream) {
    const float* x   = (const float*)d_in[0];
    const int*   ei  = (const int*)d_in[1];
    const float* Wl0 = (const float*)d_in[2];
    const float* bl0 = (const float*)d_in[3];
    const float* Wr0 = (const float*)d_in[4];
    const float* Wl1 = (const float*)d_in[5];
    const float* bl1 = (const float*)d_in[6];
    const float* Wr1 = (const float*)d_in[7];
    const float* Wl2 = (const float*)d_in[8];
    const float* bl2 = (const float*)d_in[9];
    const float* Wr2 = (const float*)d_in[10];

    const int N = in_sizes[0] / F_DIM;
    const int E = in_sizes[1] / 2;
    const int C = 4;
    const int* src = ei;
    const int* dst = ei + E;

    // workspace: deg/invdeg (N, padded) | agg (N*128) | h (N*128)
    float* ws  = (float*)d_ws;
    const size_t NPAD = ((size_t)N + 1023) & ~(size_t)1023;
    float* deg = ws;
    float* agg = ws + NPAD;
    float* h   = agg + (size_t)N * F_DIM;

    float* out0 = (float*)d_out;
    float* out1 = out0 + (size_t)N * C;
    float* h2   = out0 + (size_t)2 * N * C;   // h2 lives directly in d_out

    const long long nf = (long long)N * F_DIM;
    const int ZB = 256;
    dim3 gemmGrid((N + 15) / 16);
    dim3 gemmBlock(32, 8);
    const long long scatterThreads = (long long)E * 32;
    const int scatterBlocks = (int)((scatterThreads + ZB - 1) / ZB);

    // degree -> inv_deg (in place)
    zero_kernel<<<(int)((N + ZB - 1) / ZB), ZB, 0, stream>>>(deg, N);
    deg_kernel<<<(E + ZB - 1) / ZB, ZB, 0, stream>>>(dst, deg, E);
    invdeg_kernel<<<(N + ZB - 1) / ZB, ZB, 0, stream>>>(deg, N);

    // layer 0: h = relu(mean(x)@Wl0 + bl0 + x@Wr0)
    zero_kernel<<<(int)((nf + ZB - 1) / ZB), ZB, 0, stream>>>(agg, nf);
    scatter_kernel<<<scatterBlocks, ZB, 0, stream>>>(x, src, dst, agg, E);
    sage_gemm128_wmma<<<gemmGrid, gemmBlock, 0, stream>>>(agg, x, deg, Wl0, bl0, Wr0,
                                                          h, N, 1);

    // layer 1: h2 = mean(h)@Wl1 + bl1 + h@Wr1
    zero_kernel<<<(int)((nf + ZB - 1) / ZB), ZB, 0, stream>>>(agg, nf);
    scatter_kernel<<<scatterBlocks, ZB, 0, stream>>>(h, src, dst, agg, E);
    sage_gemm128_wmma<<<gemmGrid, gemmBlock, 0, stream>>>(agg, h, deg, Wl1, bl1, Wr1,
                                                          h2, N, 0);

    // layer 2: out = mean(h2)@Wl2 + bl2 + h2@Wr2  (written twice)
    zero_kernel<<<(int)((nf + ZB - 1) / ZB), ZB, 0, stream>>>(agg, nf);
    scatter_kernel<<<scatterBlocks, ZB, 0, stream>>>(h2, src, dst, agg, E);
    out_kernel<<<(N + ZB - 1) / ZB, ZB, 0, stream>>>(agg, h2, deg, Wl2, bl2, Wr2,
                                                     out0, out1, N);
}